// KNNModel_63917703299132
// MI455X (gfx1250) — compile-verified
//
#include <hip/hip_runtime.h>
#include <hip/hip_bf16.h>

// ---------------------------------------------------------------------------
// KNN regression: Q=4096 queries, N=50000 train, D=128, k=5.
// Ranking key per query: t_sq[n] - 2*dot(x[q], xt[n]) (x_sq & sqrt are
// rank-preserving, dropped). fp32 WMMA 16x16x4; B staged into LDS with
// CDNA5 async global->LDS copies (ASYNCcnt), double-buffered, 2 tiles/iter.
// ---------------------------------------------------------------------------

#define Q_TOTAL   4096
#define NTRAIN    50000
#define DIM       128
#define KNN       5
#define NSPLIT    16
#define ITERS_PER_SPLIT 98                        // 98 iters x 32 cols = 3136
#define COLS_PER_SPLIT  (ITERS_PER_SPLIT * 32)
#define WAVES_PER_WG 4
#define QPW       16
#define Q_PER_WG  (WAVES_PER_WG * QPW)            // 64

#define ROWSTRIDE   132                            // 128 + 4 pad (bank-conflict-free)
#define BUF_FLOATS  (32 * ROWSTRIDE)               // 4224 floats = 16896 B
#define SMEM_BYTES  40960                          // merge overlay dominates

#define TSQ_BYTES    (NTRAIN * sizeof(float))                   // 200000
#define CAND_COUNT   ((size_t)Q_TOTAL * NSPLIT * KNN)           // 327680
#define CAND_D_BYTES (CAND_COUNT * sizeof(float))
// ws layout: [tsq][cand_d][cand_i]  (~2.9 MB total)

typedef __attribute__((ext_vector_type(2))) float v2f;
typedef __attribute__((ext_vector_type(8))) float v8f;

typedef int v4i __attribute__((vector_size(16)));
typedef __attribute__((address_space(1))) v4i* gv4_ptr;   // global int4*
typedef __attribute__((address_space(3))) v4i* lv4_ptr;   // LDS int4*

#if defined(__has_builtin)
#if __has_builtin(__builtin_amdgcn_global_load_async_to_lds_b128)
#define HAVE_ASYNC_B128_BUILTIN 1
#endif
#if __has_builtin(__builtin_amdgcn_s_wait_asynccnt)
#define HAVE_WAIT_ASYNC_BUILTIN 1
#endif
#endif

// async 16B global -> LDS copy (per-lane addresses), tracked by ASYNCcnt
__device__ __forceinline__ void async_copy16(const float* gsrc, float* ldst)
{
#if defined(HAVE_ASYNC_B128_BUILTIN)
    __builtin_amdgcn_global_load_async_to_lds_b128(
        (gv4_ptr)(float*)gsrc, (lv4_ptr)ldst, 0, 0);
#else
    unsigned loff = (unsigned)(__SIZE_TYPE__)(lv4_ptr)ldst;   // LDS byte offset
    asm volatile("global_load_async_to_lds_b128 %0, %1, off"
                 :: "v"(loff), "v"(gsrc) : "memory");
#endif
}

__device__ __forceinline__ void wait_async0()
{
#if defined(HAVE_WAIT_ASYNC_BUILTIN)
    __builtin_amdgcn_s_wait_asynccnt(0);
#else
    asm volatile("s_wait_asynccnt 0x0" ::: "memory");
#endif
}

// ---------------------------------------------------------------------------
// Kernel 0: t_sq[n] = sum_k xt[n][k]^2   (one wave32 per row, float4 loads)
// ---------------------------------------------------------------------------
__global__ __launch_bounds__(256) void tsq_kernel(const float* __restrict__ xt,
                                                  float* __restrict__ tsq)
{
    const int wave = threadIdx.x >> 5;
    const int lane = threadIdx.x & 31;
    const int row  = blockIdx.x * 8 + wave;
    if (row >= NTRAIN) return;
    const float4* p = (const float4*)(xt + (size_t)row * DIM);
    float4 v = p[lane];
    float s = v.x * v.x + v.y * v.y + v.z * v.z + v.w * v.w;
    s += __shfl_xor(s, 1, 32);
    s += __shfl_xor(s, 2, 32);
    s += __shfl_xor(s, 4, 32);
    s += __shfl_xor(s, 8, 32);
    s += __shfl_xor(s, 16, 32);
    if (lane == 0) tsq[row] = s;
}

// ---------------------------------------------------------------------------
// Kernel 1: fused fp32-WMMA distance + per-query top-5 within an N-split.
// ---------------------------------------------------------------------------
__global__ __launch_bounds__(128) void knn_tile_kernel(
    const float* __restrict__ x, const float* __restrict__ xt,
    const float* __restrict__ tsq,
    float* __restrict__ cand_d, int* __restrict__ cand_i)
{
    __shared__ __align__(16) char smem[SMEM_BYTES];
    float* smemF = (float*)smem;                      // stage buffers (loop)
    float (*sd)[32][40] = (float(*)[32][40])smem;     // merge overlay (post-loop)
    int   (*si)[32][40] = (int(*)[32][40])(smem + 20480);

    const int tid   = threadIdx.x;
    const int wave  = tid >> 5;
    const int lane  = tid & 31;
    const int m     = lane & 15;                 // A row / B column within tile
    const int kk    = (lane < 16) ? 0 : 2;       // K sub-offset per ISA layout
    const int qbase = blockIdx.x * Q_PER_WG + wave * QPW;
    const int split = blockIdx.y;
    const int col0  = split * COLS_PER_SPLIT;

    // issue async stage of one 32-row (16 KB) block into LDS buffer `buf`.
    // 128 threads x 8 chunks x 16 B; per-lane LDS scatter handles row padding.
    auto issue_stage = [&](int it, int buf) {
#pragma unroll
        for (int p = 0; p < 8; ++p) {
            const int f   = tid * 4 + p * 512;       // float index in 32x128 block
            const int row = f >> 7;
            const int ci  = f & 127;
            int grow = col0 + it * 32 + row;
            grow = (grow < NTRAIN) ? grow : (NTRAIN - 1);    // clamp OOB rows
            async_copy16(xt + (size_t)grow * DIM + ci,
                         smemF + buf * BUF_FLOATS + row * ROWSTRIDE + ci);
        }
    };

    issue_stage(0, 0);                 // ASYNCcnt stream starts immediately

    // ---- A panel: 16 queries x 128 dims fp32 resident in 64 VGPRs ---------
    // (LOADcnt; independent of the ASYNCcnt staging stream)
    v2f a[32];
    {
        const float* ap = x + (size_t)(qbase + m) * DIM + kk;
#pragma unroll
        for (int kf = 0; kf < 32; ++kf)
            a[kf] = *(const v2f*)(ap + kf * 4);
    }

    // ---- per-lane top-5 lists for the 8 C rows this lane holds ------------
    float td[8][KNN];
    int   ti[8][KNN];
#pragma unroll
    for (int r = 0; r < 8; ++r)
#pragma unroll
        for (int j = 0; j < KNN; ++j) { td[r][j] = __builtin_inff(); ti[r][j] = 0; }

    wait_async0();
    __syncthreads();                   // buffer 0 ready for all waves
    int cur = 0;

    for (int it = 0; it < ITERS_PER_SPLIT; ++it) {
        if (it + 1 < ITERS_PER_SPLIT)
            issue_stage(it + 1, cur ^ 1);            // fills other buffer

        // per-lane columns for the two 16-wide tiles of this iteration
        const int colA  = col0 + it * 32 + m;
        const int colB  = colA + 16;
        const int colAc = (colA < NTRAIN) ? colA : (NTRAIN - 1);
        const int colBc = (colB < NTRAIN) ? colB : (NTRAIN - 1);
        const float ts0 = tsq[colAc];                // LOADcnt, cheap
        const float ts1 = tsq[colBc];

        // ---- 64x WMMA: two interleaved accumulator chains -----------------
        const float* l0 = smemF + cur * BUF_FLOATS + m * ROWSTRIDE + kk;
        const float* l1 = l0 + 16 * ROWSTRIDE;
        v8f c0 = {0.f, 0.f, 0.f, 0.f, 0.f, 0.f, 0.f, 0.f};
        v8f c1 = {0.f, 0.f, 0.f, 0.f, 0.f, 0.f, 0.f, 0.f};
#pragma unroll
        for (int bt = 0; bt < 8; ++bt) {
            v2f b0[4], b1[4];
#pragma unroll
            for (int i = 0; i < 4; ++i) {
                b0[i] = *(const v2f*)(l0 + (bt * 4 + i) * 4);
                b1[i] = *(const v2f*)(l1 + (bt * 4 + i) * 4);
            }
#pragma unroll
            for (int i = 0; i < 4; ++i) {
                c0 = __builtin_amdgcn_wmma_f32_16x16x4_f32(
                        false, a[bt * 4 + i], false, b0[i], (short)0, c0, false, false);
                c1 = __builtin_amdgcn_wmma_f32_16x16x4_f32(
                        false, a[bt * 4 + i], false, b1[i], (short)0, c1, false, false);
            }
        }

        const bool vA = (colA < NTRAIN);
        const bool vB = (colB < NTRAIN);
#pragma unroll
        for (int r = 0; r < 8; ++r) {
#pragma unroll
            for (int half = 0; half < 2; ++half) {
                float key = half ? (vB ? fmaf(-2.0f, c1[r], ts1) : __builtin_inff())
                                 : (vA ? fmaf(-2.0f, c0[r], ts0) : __builtin_inff());
                int   idx = half ? colB : colA;
                if (key < td[r][KNN - 1]) {                  // rare insert
                    td[r][KNN - 1] = key; ti[r][KNN - 1] = idx;
#pragma unroll
                    for (int j = KNN - 1; j > 0; --j) {
                        bool  sw = td[r][j] < td[r][j - 1];
                        float d0 = td[r][j - 1]; int i0 = ti[r][j - 1];
                        td[r][j - 1] = sw ? td[r][j] : d0;
                        ti[r][j - 1] = sw ? ti[r][j] : i0;
                        td[r][j]     = sw ? d0 : td[r][j];
                        ti[r][j]     = sw ? i0 : ti[r][j];
                    }
                }
            }
        }

        wait_async0();                 // own async writes for it+1 complete
        __syncthreads();               // all waves: writes visible, reads done
        cur ^= 1;
    }

    // ---- dump lane lists to LDS (overlaying stage buffers), exact merge ----
#pragma unroll
    for (int r = 0; r < 8; ++r)
#pragma unroll
        for (int j = 0; j < KNN; ++j) {
            sd[wave][lane][r * KNN + j] = td[r][j];
            si[wave][lane][r * KNN + j] = ti[r][j];
        }
    __syncthreads();

    if (lane < 16) {
        const int row    = lane;                      // query row 0..15
        const int lbase2 = (row < 8) ? 0 : 16;        // lane half holding it
        const int slot   = (row & 7) * KNN;
        float bd[KNN]; int bi[KNN];
#pragma unroll
        for (int j = 0; j < KNN; ++j) { bd[j] = __builtin_inff(); bi[j] = 0; }
        for (int l = 0; l < 16; ++l) {
#pragma unroll
            for (int j = 0; j < KNN; ++j) {
                float d  = sd[wave][lbase2 + l][slot + j];
                int   id = si[wave][lbase2 + l][slot + j];
                if (d < bd[KNN - 1]) {
                    bd[KNN - 1] = d; bi[KNN - 1] = id;
#pragma unroll
                    for (int jj = KNN - 1; jj > 0; --jj) {
                        bool  sw = bd[jj] < bd[jj - 1];
                        float d0 = bd[jj - 1]; int i0 = bi[jj - 1];
                        bd[jj - 1] = sw ? bd[jj] : d0;
                        bi[jj - 1] = sw ? bi[jj] : i0;
                        bd[jj]     = sw ? d0 : bd[jj];
                        bi[jj]     = sw ? i0 : bi[jj];
                    }
                }
            }
        }
        const int q = qbase + row;
        const size_t base = ((size_t)q * NSPLIT + split) * KNN;
#pragma unroll
        for (int j = 0; j < KNN; ++j) {
            cand_d[base + j] = bd[j];
            cand_i[base + j] = bi[j];
        }
    }
}

// ---------------------------------------------------------------------------
// Kernel 2: merge 16 splits x 5 candidates per query, gather y, mean.
// ---------------------------------------------------------------------------
__global__ __launch_bounds__(256) void knn_merge_kernel(
    const float* __restrict__ cand_d, const int* __restrict__ cand_i,
    const float* __restrict__ y, float* __restrict__ out)
{
    const int q = blockIdx.x * 256 + threadIdx.x;
    if (q >= Q_TOTAL) return;

    float bd[KNN]; int bi[KNN];
#pragma unroll
    for (int j = 0; j < KNN; ++j) { bd[j] = __builtin_inff(); bi[j] = 0; }

    const float* cd = cand_d + (size_t)q * NSPLIT * KNN;
    const int*   ci = cand_i + (size_t)q * NSPLIT * KNN;
    for (int s = 0; s < NSPLIT * KNN; ++s) {
        float d  = cd[s];
        int   id = ci[s];
        if (d < bd[KNN - 1]) {
            bd[KNN - 1] = d; bi[KNN - 1] = id;
#pragma unroll
            for (int j = KNN - 1; j > 0; --j) {
                bool  sw = bd[j] < bd[j - 1];
                float d0 = bd[j - 1]; int i0 = bi[j - 1];
                bd[j - 1] = sw ? bd[j] : d0;
                bi[j - 1] = sw ? bi[j] : i0;
                bd[j]     = sw ? d0 : bd[j];
                bi[j]     = sw ? i0 : bi[j];
            }
        }
    }

    float acc = 0.0f;
#pragma unroll
    for (int j = 0; j < KNN; ++j) {
        int id = bi[j];
        id = (id < 0) ? 0 : ((id >= NTRAIN) ? (NTRAIN - 1) : id);
        acc += y[id];
    }
    out[q] = acc * (1.0f / KNN);
}

// ---------------------------------------------------------------------------
extern "C" void kernel_launch(void* const* d_in, const int* in_sizes, int n_in,
                              void* d_out, int out_size, void* d_ws, size_t ws_size,
                              hipStream_t stream)
{
    const float* x  = (const float*)d_in[0];   // [4096,128]
    const float* xt = (const float*)d_in[1];   // [50000,128]
    const float* y  = (const float*)d_in[2];   // [50000,1]
    float* out = (float*)d_out;                // [4096,1]

    float* tsq    = (float*)d_ws;
    float* cand_d = (float*)((char*)d_ws + TSQ_BYTES);
    int*   cand_i = (int*)((char*)d_ws + TSQ_BYTES + CAND_D_BYTES);

    tsq_kernel<<<6250, 256, 0, stream>>>(xt, tsq);

    dim3 grid(Q_TOTAL / Q_PER_WG, NSPLIT);
    knn_tile_kernel<<<grid, WAVES_PER_WG * 32, 0, stream>>>(x, xt, tsq, cand_d, cand_i);

    knn_merge_kernel<<<Q_TOTAL / 256, 256, 0, stream>>>(cand_d, cand_i, y, out);
}